// GCN_43576738186087
// MI455X (gfx1250) — compile-verified
//
#include <hip/hip_runtime.h>
#include <hip/hip_bf16.h>

typedef __attribute__((ext_vector_type(2))) float v2f;
typedef __attribute__((ext_vector_type(8))) float v8f;

#define N_NODES   50000
#define N_EDGES   800000
#define FEATS     128
#define OUTF      16
#define NUM_G     128

// ---------------- degree / normalization ----------------
__global__ void init_deg_kernel(float* __restrict__ deg) {
    int i = blockIdx.x * blockDim.x + threadIdx.x;
    if (i < N_NODES) deg[i] = 1.0f;                 // self-loop counts once
}

__global__ void accum_deg_kernel(const int* __restrict__ dst, float* __restrict__ deg) {
    int e = blockIdx.x * blockDim.x + threadIdx.x;
    if (e < N_EDGES) atomicAdd(&deg[dst[e]], 1.0f);
}

__global__ void finalize_dinv_kernel(float* __restrict__ deg) {
    int i = blockIdx.x * blockDim.x + threadIdx.x;
    if (i < N_NODES) deg[i] = rsqrtf(deg[i]);       // deg >= 1 always (self-loops)
}

// ---------------- fp32 WMMA GEMM: C[M,128] = A[M,128] @ W[128,128] ----------------
// grid.x = M/16 (M == 50000 -> 3125), block = 256 (8 waves), wave w owns N-tile w.
__global__ __launch_bounds__(256) void gemm_wmma_kernel(
    const float* __restrict__ A, const float* __restrict__ W, float* __restrict__ C) {
    __shared__ float As[16][FEATS + 4];             // +4 pad: kills bank conflicts
    const int tid   = threadIdx.x;
    const int wave  = tid >> 5;
    const int lane  = tid & 31;
    const int mbase = blockIdx.x * 16;

    // cooperative load of the 16x128 A tile (512 float4, 256 threads -> 2 each)
    {
        const float4* src4 = (const float4*)(A + (size_t)mbase * FEATS);
        for (int i = tid; i < 16 * (FEATS / 4); i += 256) {
            int r  = i >> 5;        // /32
            int c4 = i & 31;
            float4 v = src4[r * (FEATS / 4) + c4];
            As[r][c4 * 4 + 0] = v.x;
            As[r][c4 * 4 + 1] = v.y;
            As[r][c4 * 4 + 2] = v.z;
            As[r][c4 * 4 + 3] = v.w;
        }
    }
    __syncthreads();

    const int mrow  = lane & 15;                    // A row within tile
    const int nc    = (wave << 4) + (lane & 15);    // global output column
    const int khalf = (lane >> 4) << 1;             // 0 or 2

    v8f acc = {};
#pragma unroll
    for (int k = 0; k < FEATS; k += 4) {
        const int kb = k + khalf;
        v2f a;                                      // 16x4 fp32 A-frag layout
        a.x = As[mrow][kb];
        a.y = As[mrow][kb + 1];
        v2f b;                                      // 4x16 fp32 B-frag layout
        b.x = W[(size_t)kb * FEATS + nc];
        b.y = W[(size_t)(kb + 1) * FEATS + nc];
        acc = __builtin_amdgcn_wmma_f32_16x16x4_f32(
            /*neg_a=*/false, a, /*neg_b=*/false, b,
            /*c_mod=*/(short)0, acc, /*reuse_a=*/false, /*reuse_b=*/false);
    }

    // D layout: VGPR v -> row (v + 8*(lane>=16)), col = lane&15 within tile
    const int rbase = mbase + ((lane >> 4) << 3);
#pragma unroll
    for (int v = 0; v < 8; v++)
        C[(size_t)(rbase + v) * FEATS + nc] = acc[v];
}

// ---------------- message passing ----------------
// agg = xw * dinv^2  (self-loop contribution; also serves as the zero-init)
__global__ void init_agg_kernel(const float* __restrict__ xw, const float* __restrict__ dinv,
                                float* __restrict__ agg) {
    long idx = (long)blockIdx.x * blockDim.x + threadIdx.x;
    if (idx < (long)N_NODES * FEATS) {
        float di = dinv[idx >> 7];
        agg[idx] = xw[idx] * di * di;
    }
}

// one wave per edge: float4 gather from L2-resident xw, 4 L2 atomics per lane
__global__ __launch_bounds__(256) void edge_scatter_kernel(
    const float* __restrict__ xw, const float* __restrict__ dinv,
    const int* __restrict__ src, const int* __restrict__ dst, float* __restrict__ agg) {
    int e = blockIdx.x * (blockDim.x >> 5) + (threadIdx.x >> 5);
    if (e >= N_EDGES) return;
    const int lane = threadIdx.x & 31;
    const int s = src[e];
    const int d = dst[e];
    const float w = dinv[s] * dinv[d];
    const float4 v = ((const float4*)(xw + (size_t)s * FEATS))[lane];
    float* o = agg + (size_t)d * FEATS + lane * 4;
    atomicAdd(o + 0, v.x * w);
    atomicAdd(o + 1, v.y * w);
    atomicAdd(o + 2, v.z * w);
    atomicAdd(o + 3, v.w * w);
}

__global__ void bias_relu_kernel(float* __restrict__ h, const float* __restrict__ b) {
    long idx = (long)blockIdx.x * blockDim.x + threadIdx.x;
    if (idx < (long)N_NODES * FEATS) {
        float v = h[idx] + b[idx & (FEATS - 1)];
        h[idx] = v > 0.0f ? v : 0.0f;
    }
}

// ---------------- pooling + head ----------------
__global__ void zero_pool_kernel(float* __restrict__ pooled, float* __restrict__ counts) {
    int i = blockIdx.x * blockDim.x + threadIdx.x;
    if (i < NUM_G * FEATS) pooled[i] = 0.0f;
    if (i < NUM_G) counts[i] = 0.0f;
}

__global__ __launch_bounds__(256) void pool_accum_kernel(
    const float* __restrict__ h, const int* __restrict__ batch,
    float* __restrict__ pooled, float* __restrict__ counts) {
    int node = blockIdx.x * (blockDim.x >> 5) + (threadIdx.x >> 5);
    if (node >= N_NODES) return;
    const int lane = threadIdx.x & 31;
    const int g = batch[node];
    const float4 v = ((const float4*)(h + (size_t)node * FEATS))[lane];
    float* o = pooled + (size_t)g * FEATS + lane * 4;
    atomicAdd(o + 0, v.x);
    atomicAdd(o + 1, v.y);
    atomicAdd(o + 2, v.z);
    atomicAdd(o + 3, v.w);
    if (lane == 0) atomicAdd(&counts[g], 1.0f);
}

__global__ void pool_final_kernel(const float* __restrict__ pooled, const float* __restrict__ counts,
                                  const float* __restrict__ lin_w, const float* __restrict__ lin_b,
                                  float* __restrict__ out) {
    __shared__ float s[FEATS];
    const int g = blockIdx.x;
    const int t = threadIdx.x;                      // 128 threads
    const float inv = 1.0f / fmaxf(counts[g], 1.0f);
    s[t] = pooled[(size_t)g * FEATS + t] * inv;
    __syncthreads();
    if (t < OUTF) {
        float acc = lin_b[t];
#pragma unroll 8
        for (int f = 0; f < FEATS; f++) acc += s[f] * lin_w[f * OUTF + t];
        out[(size_t)g * OUTF + t] = acc;
    }
}

// ---------------- host orchestration ----------------
extern "C" void kernel_launch(void* const* d_in, const int* in_sizes, int n_in,
                              void* d_out, int out_size, void* d_ws, size_t ws_size,
                              hipStream_t stream) {
    const float* x     = (const float*)d_in[0];
    const int*   edge  = (const int*)d_in[1];      // [2, E] flat: row0=src, row1=dst
    const int*   batch = (const int*)d_in[2];
    const float* W0    = (const float*)d_in[3];
    const float* b0    = (const float*)d_in[4];
    const float* W1    = (const float*)d_in[5];
    const float* b1    = (const float*)d_in[6];
    const float* W2    = (const float*)d_in[7];
    const float* b2    = (const float*)d_in[8];
    const float* lin_w = (const float*)d_in[9];
    const float* lin_b = (const float*)d_in[10];
    float* out = (float*)d_out;

    const int* src = edge;
    const int* dst = edge + N_EDGES;

    // workspace carve-up (all 256B-aligned)
    char* ws = (char*)d_ws;
    float* dinv   = (float*)ws;                               // 50000 floats
    float* bufA   = (float*)(ws + 200704);                    // 6.4M floats (xw)
    float* bufB   = bufA + (size_t)N_NODES * FEATS;           // 6.4M floats (agg/h)
    float* pooled = bufB + (size_t)N_NODES * FEATS;           // 16384 floats
    float* counts = pooled + NUM_G * FEATS;                   // 128 floats

    const int T = 256;
    const long NF = (long)N_NODES * FEATS;

    // symmetric degree normalization
    init_deg_kernel<<<(N_NODES + T - 1) / T, T, 0, stream>>>(dinv);
    accum_deg_kernel<<<(N_EDGES + T - 1) / T, T, 0, stream>>>(dst, dinv);
    finalize_dinv_kernel<<<(N_NODES + T - 1) / T, T, 0, stream>>>(dinv);

    const float* Ws[3] = {W0, W1, W2};
    const float* bs[3] = {b0, b1, b2};
    const float* hin = x;
    for (int l = 0; l < 3; l++) {
        gemm_wmma_kernel<<<N_NODES / 16, 256, 0, stream>>>(hin, Ws[l], bufA);
        init_agg_kernel<<<(int)((NF + T - 1) / T), T, 0, stream>>>(bufA, dinv, bufB);
        edge_scatter_kernel<<<(N_EDGES + 7) / 8, 256, 0, stream>>>(bufA, dinv, src, dst, bufB);
        bias_relu_kernel<<<(int)((NF + T - 1) / T), T, 0, stream>>>(bufB, bs[l]);
        hin = bufB;
    }

    // global mean pool + linear head
    zero_pool_kernel<<<(NUM_G * FEATS + T - 1) / T, T, 0, stream>>>(pooled, counts);
    pool_accum_kernel<<<(N_NODES + 7) / 8, 256, 0, stream>>>(bufB, batch, pooled, counts);
    pool_final_kernel<<<NUM_G, FEATS, 0, stream>>>(pooled, counts, lin_w, lin_b, out);
}